// GraphModel_33578054320281
// MI455X (gfx1250) — compile-verified
//
#include <hip/hip_runtime.h>
#include <hip/hip_bf16.h>
#include <math.h>

typedef float v2f __attribute__((ext_vector_type(2)));
typedef float v8f __attribute__((ext_vector_type(8)));

static constexpr int Bb = 4, Tt = 12, Nn = 50000, Fin = 8, Hh = 32, Kk = 3;
static constexpr int Mrows = Bb * Nn; // 200000, multiple of 16
static constexpr int ScanB = (Nn + 255) / 256; // 196 (<=256, fits one 2nd-level block)

// ---------------------------------------------------------------- utilities

__global__ void k_zero(float* __restrict__ p, long n) {
  long i = (long)blockIdx.x * blockDim.x + threadIdx.x;
  if (i < n) p[i] = 0.f;
}

__global__ void k_zero_i(int* __restrict__ p, long n) {
  long i = (long)blockIdx.x * blockDim.x + threadIdx.x;
  if (i < n) p[i] = 0;
}

// deg[row] += w  (weighted degree, matches segment_sum over row)
__global__ void k_deg(const int* __restrict__ row, const float* __restrict__ w,
                      float* __restrict__ deg, int E) {
  int e = blockIdx.x * blockDim.x + threadIdx.x;
  if (e < E) atomicAdd(&deg[row[e]], w[e]);
}

// deg -> dinv (in place), diag = (deg>0 ? 0 : -1)
__global__ void k_dinv_diag(float* __restrict__ deg, float* __restrict__ diag, int N) {
  int i = blockIdx.x * blockDim.x + threadIdx.x;
  if (i >= N) return;
  float d = deg[i];
  diag[i] = d > 0.f ? 0.f : -1.f;
  deg[i]  = d > 0.f ? rsqrtf(d) : 0.f;
}

// norm[e] = -w * dinv[row] * dinv[col]
__global__ void k_norm(const int* __restrict__ row, const int* __restrict__ col,
                       const float* __restrict__ w, const float* __restrict__ dinv,
                       float* __restrict__ norm, int E) {
  int e = blockIdx.x * blockDim.x + threadIdx.x;
  if (e < E) norm[e] = -w[e] * dinv[row[e]] * dinv[col[e]];
}

// ---------------------------------------------- CSC build (gather layout)

__global__ void k_count(const int* __restrict__ col, int* __restrict__ cnt, int E) {
  int e = blockIdx.x * blockDim.x + threadIdx.x;
  if (e < E) atomicAdd(&cnt[col[e]], 1);
}

// per-block inclusive scan of cnt -> off (local), block totals -> bsum
__global__ void k_scan_block(const int* __restrict__ cnt, int* __restrict__ off,
                             int* __restrict__ bsum, int N) {
  __shared__ int s[256];
  int t = threadIdx.x;
  int i = blockIdx.x * 256 + t;
  s[t] = (i < N) ? cnt[i] : 0;
  __syncthreads();
  for (int d = 1; d < 256; d <<= 1) {
    int x = (t >= d) ? s[t - d] : 0;
    __syncthreads();
    s[t] += x;
    __syncthreads();
  }
  if (i < N) off[i] = s[t];
  if (t == 255) bsum[blockIdx.x] = s[255];
}

__global__ void k_scan_bsums(int* __restrict__ bsum, int nB) {
  __shared__ int s[256];
  int t = threadIdx.x;
  s[t] = (t < nB) ? bsum[t] : 0;
  __syncthreads();
  for (int d = 1; d < 256; d <<= 1) {
    int x = (t >= d) ? s[t - d] : 0;
    __syncthreads();
    s[t] += x;
    __syncthreads();
  }
  if (t < nB) bsum[t] = s[t];
}

// off[i] -> exclusive global offset; cur[i] = off[i]; off[N] = E
__global__ void k_scan_final(const int* __restrict__ cnt, int* __restrict__ off,
                             const int* __restrict__ bsum, int* __restrict__ cur,
                             int N, int E) {
  int i = blockIdx.x * blockDim.x + threadIdx.x;
  if (i == 0) off[N] = E;
  if (i >= N) return;
  int bid = i >> 8;
  int prev = (bid > 0) ? bsum[bid - 1] : 0;
  int v = off[i] - cnt[i] + prev; // exclusive
  off[i] = v;
  cur[i] = v;
}

// edge record = {src_row, norm} packed to one 8B load
__global__ void k_place(const int* __restrict__ row, const int* __restrict__ col,
                        const float* __restrict__ norm, int* __restrict__ cur,
                        int2* __restrict__ sedge, int E) {
  int e = blockIdx.x * blockDim.x + threadIdx.x;
  if (e >= E) return;
  int p = atomicAdd(&cur[col[e]], 1);
  sedge[p] = make_int2(row[e], __float_as_int(norm[e]));
}

// ------------------------------------------------- Chebyshev basis kernels
// Tbuf layout: [b, n, k, f] i.e. node row = [T0(F) | T1(F) | T2(F)], stride 3F.

// T0 = x
__global__ void k_copy0(const float* __restrict__ x, long xbstride,
                        float* __restrict__ T, int F, int total) {
  int i = blockIdx.x * blockDim.x + threadIdx.x;
  if (i >= total) return;
  int f = i % F; int bn = i / F; int b = bn / Nn; int n = bn - b * Nn;
  T[(size_t)bn * 3 * F + f] = x[(long)b * xbstride + (long)n * F + f];
}

// Fused gather SpMM + Chebyshev epilogue (atomic-free).
// MODE 1: T1[n] = sum_j w_j * x[r_j] + diag[n]*x[n]          (reads x, stride F)
// MODE 2: T2[n] = 2*(sum_j w_j * T1[r_j] + diag[n]*T1[n]) - T0[n]
// F lanes cover one node; at F=32 one wave32 == one node, node id is forced
// through readfirstlane so edge-record/offset loads take the scalar path and
// the vector pipe only carries the coalesced 128B feature-row gathers.
template <int F, int MODE>
__global__ void k_gather(const float* __restrict__ xsrc, long xb,
                         float* __restrict__ Tbuf, const int2* __restrict__ sedge,
                         const int* __restrict__ off, const float* __restrict__ diag) {
  int n;
  if constexpr (F == 32) {
    n = __builtin_amdgcn_readfirstlane(blockIdx.x * 8 + (threadIdx.x >> 5));
  } else {
    n = blockIdx.x * (256 / F) + threadIdx.x / F;
  }
  int f = threadIdx.x & (F - 1);
  int b = blockIdx.y;
  if (n >= Nn) return;
  int j0 = off[n], j1 = off[n + 1];
  float acc = 0.f;
  float* base = Tbuf + (size_t)(b * Nn + n) * (3 * F);
  if constexpr (MODE == 1) {
    const float* src = xsrc + (long)b * xb;
    for (int j = j0; j < j1; ++j) {
      int2 e = sedge[j];
      acc += __int_as_float(e.y) * src[(size_t)e.x * F + f];
    }
    base[F + f] = acc + diag[n] * src[(size_t)n * F + f];
  } else {
    const float* t1 = Tbuf + (size_t)b * Nn * (3 * F) + F;
    for (int j = j0; j < j1; ++j) {
      int2 e = sedge[j];
      acc += __int_as_float(e.y) * t1[(size_t)e.x * (3 * F) + f];
    }
    base[2 * F + f] = 2.f * (acc + diag[n] * base[F + f]) - base[f];
  }
}

// ------------------------------------------------------------ WMMA GEMMs
// f32 16x16x4 WMMA (full f32 precision, CDNA5 wave32 matrix pipe).
// A (16x4): lanes 0-15 hold K={0,1} in {v0,v1}; lanes 16-31 hold K={2,3}.
// B (4x16): lane (n = lane&15); v0 = B[K=0|2][n], v1 = B[K=1|3][n].
// C/D (16x16, 8 VGPRs): VGPR j -> M = j + 8*(lane>=16), N = lane&15.

static __device__ __forceinline__ v8f wmma_f32(v2f a, v2f b, v8f c) {
  return __builtin_amdgcn_wmma_f32_16x16x4_f32(false, a, false, b,
                                               (short)0, c, false, false);
}

static __device__ __forceinline__ void load_Wtile(const float* __restrict__ W,
                                                  float* __restrict__ Wl,
                                                  int KK, int NN, int nTile,
                                                  int tid, int nthreads) {
  for (int i = tid; i < KK * 16; i += nthreads) {
    int k = i >> 4, n = i & 15;
    Wl[i] = W[(size_t)k * NN + nTile + n];
  }
}

static __device__ __forceinline__ v8f gemm_tile(const float* __restrict__ A,
                                                const float* __restrict__ Wl,
                                                int KK, int mTile, int lane) {
  int m = lane & 15, hi = lane >> 4;
  const float* Arow = A + (size_t)(mTile + m) * KK;
  v8f acc = 0.0f;
  for (int k = 0; k < KK; k += 4) {
    v2f a, b;
    a.x = Arow[k + 2 * hi + 0];
    a.y = Arow[k + 2 * hi + 1];
    b.x = Wl[(k + 2 * hi + 0) * 16 + m];
    b.y = Wl[(k + 2 * hi + 1) * 16 + m];
    acc = wmma_f32(a, b, acc);
  }
  return acc;
}

// C[M,NN] = A[M,KK] @ W[KK,NN] + bias[NN]   (gx, NN=96, bias = bx)
__global__ void k_gemm_bias(const float* __restrict__ A, const float* __restrict__ W,
                            const float* __restrict__ bias, float* __restrict__ C,
                            int M, int KK, int NN) {
  __shared__ float Wl[96 * 16];
  int nTile = blockIdx.y * 16;
  load_Wtile(W, Wl, KK, NN, nTile, threadIdx.x, blockDim.x);
  __syncthreads();
  int wave = threadIdx.x >> 5, lane = threadIdx.x & 31;
  int mTile = blockIdx.x * 128 + wave * 16;
  if (mTile >= M) return;
  v8f acc = gemm_tile(A, Wl, KK, mTile, lane);
  int m = lane & 15, hi = lane >> 4;
  float bv = bias[nTile + m];
  for (int j = 0; j < 8; ++j) {
    int r = mTile + j + hi * 8;
    C[(size_t)r * NN + nTile + m] = acc[j] + bv;
  }
}

// z/r gates: acc = TH @ Wzr[96,64]; col<32 -> Z=sigmoid(acc+bh_z+gx0) stored
// in-place over gx slot 0; col>=32 -> R=sigmoid(acc+bh_r+gx1), HR = H*R.
__global__ void k_gates_zr(const float* __restrict__ TH, const float* __restrict__ Wzr,
                           const float* __restrict__ bh, float* __restrict__ gx,
                           const float* __restrict__ H, float* __restrict__ HR, int M) {
  __shared__ float Wl[96 * 16];
  int nTile = blockIdx.y * 16;
  load_Wtile(Wzr, Wl, 96, 64, nTile, threadIdx.x, blockDim.x);
  __syncthreads();
  int wave = threadIdx.x >> 5, lane = threadIdx.x & 31;
  int mTile = blockIdx.x * 128 + wave * 16;
  if (mTile >= M) return;
  v8f acc = gemm_tile(TH, Wl, 96, mTile, lane);
  int m = lane & 15, hi = lane >> 4;
  int colg = nTile + m;
  if (colg < 32) {
    float bb = bh[colg];
    for (int j = 0; j < 8; ++j) {
      int r = mTile + j + hi * 8;
      float z = 1.f / (1.f + __expf(-(acc[j] + bb + gx[(size_t)r * 96 + colg])));
      gx[(size_t)r * 96 + colg] = z;
    }
  } else {
    int hc = colg - 32;
    float bb = bh[32 + hc];
    for (int j = 0; j < 8; ++j) {
      int r = mTile + j + hi * 8;
      float rr = 1.f / (1.f + __expf(-(acc[j] + bb + gx[(size_t)r * 96 + 32 + hc])));
      HR[(size_t)r * 32 + hc] = H[(size_t)r * 32 + hc] * rr;
    }
  }
}

// candidate gate + GRU update: acc = THR @ W2[96,32];
// Ht = tanh(acc + bh2 + gx2); H = Z*H + (1-Z)*Ht  (Z lives in gx slot 0)
__global__ void k_gates_h(const float* __restrict__ THR, const float* __restrict__ W2,
                          const float* __restrict__ bh, const float* __restrict__ gx,
                          float* __restrict__ H, int M) {
  __shared__ float Wl[96 * 16];
  int nTile = blockIdx.y * 16;
  load_Wtile(W2, Wl, 96, 32, nTile, threadIdx.x, blockDim.x);
  __syncthreads();
  int wave = threadIdx.x >> 5, lane = threadIdx.x & 31;
  int mTile = blockIdx.x * 128 + wave * 16;
  if (mTile >= M) return;
  v8f acc = gemm_tile(THR, Wl, 96, mTile, lane);
  int m = lane & 15, hi = lane >> 4;
  int col = nTile + m;
  float bb = bh[64 + col];
  for (int j = 0; j < 8; ++j) {
    int r = mTile + j + hi * 8;
    float ht = tanhf(acc[j] + bb + gx[(size_t)r * 96 + 64 + col]);
    float z  = gx[(size_t)r * 96 + col];
    H[(size_t)r * 32 + col] = z * H[(size_t)r * 32 + col] + (1.f - z) * ht;
  }
}

// ------------------------------------------------------- weight repacking

__global__ void k_pack_Wx(const float* __restrict__ Wx, float* __restrict__ out, int F) {
  int i = blockIdx.x * blockDim.x + threadIdx.x;
  int total = 3 * Kk * F * Hh;
  if (i >= total) return;
  int h = i % Hh; int f = (i / Hh) % F; int k = (i / (Hh * F)) % Kk; int g = i / (Hh * F * Kk);
  out[(k * F + f) * (3 * Hh) + g * Hh + h] = Wx[((g * Kk + k) * F + f) * Hh + h];
}

__global__ void k_pack_Wh(const float* __restrict__ Wh, float* __restrict__ zr,
                          float* __restrict__ w2) {
  int i = blockIdx.x * blockDim.x + threadIdx.x;
  int total = 3 * Kk * Hh * Hh;
  if (i >= total) return;
  int h = i % Hh; int r = (i / Hh) % Hh; int k = (i / (Hh * Hh)) % Kk; int g = i / (Hh * Hh * Kk);
  float v = Wh[((g * Kk + k) * Hh + r) * Hh + h];
  if (g < 2) zr[(k * Hh + r) * 64 + g * Hh + h] = v;
  else       w2[(k * Hh + r) * 32 + h] = v;
}

// ---------------------------------------------------------------- head

__global__ void k_head(const float* __restrict__ H2, const float* __restrict__ muW,
                       const float* __restrict__ mub, const float* __restrict__ sgW,
                       const float* __restrict__ sgb, float* __restrict__ mu,
                       float* __restrict__ sg, int BN) {
  int r = blockIdx.x * blockDim.x + threadIdx.x;
  if (r >= BN) return;
  const float* h = H2 + (size_t)r * 32;
  for (int c = 0; c < 2; ++c) {
    float am = mub[c], as = sgb[c];
    for (int k = 0; k < 32; ++k) { float hv = h[k]; am += hv * muW[k * 2 + c]; as += hv * sgW[k * 2 + c]; }
    mu[(size_t)r * 2 + c] = 1.f / (1.f + __expf(-am));
    sg[(size_t)r * 2 + c] = as > 20.f ? as : log1pf(__expf(as));
  }
}

__global__ void k_mix_reduce(const float* __restrict__ H2, float* __restrict__ tmp, long total) {
  __shared__ float sb[128];
  if (threadIdx.x < 128) sb[threadIdx.x] = 0.f;
  __syncthreads();
  long stride = (long)gridDim.x * blockDim.x; // multiple of 32 -> h invariant
  int h = threadIdx.x & 31;
  for (long i = (long)blockIdx.x * blockDim.x + threadIdx.x; i < total; i += stride) {
    int b = (int)(i / (32L * Nn));
    atomicAdd(&sb[b * 32 + h], H2[i]);
  }
  __syncthreads();
  if (threadIdx.x < 128) atomicAdd(&tmp[threadIdx.x], sb[threadIdx.x] * (1.f / (float)Nn));
}

__global__ void k_mix_softmax(const float* __restrict__ tmp, float* __restrict__ out) {
  int b = threadIdx.x;
  if (b >= Bb) return;
  const float* v = tmp + b * 32;
  float mx = v[0];
  for (int i = 1; i < 32; ++i) mx = fmaxf(mx, v[i]);
  float e[32]; float s = 0.f;
  for (int i = 0; i < 32; ++i) { e[i] = __expf(v[i] - mx); s += e[i]; }
  for (int i = 0; i < 32; ++i) out[b * 32 + i] = e[i] / s;
}

// ---------------------------------------------------------------- driver

extern "C" void kernel_launch(void* const* d_in, const int* in_sizes, int n_in,
                              void* d_out, int out_size, void* d_ws, size_t ws_size,
                              hipStream_t stream) {
  const float* in   = (const float*)d_in[0];
  const int*   eidx = (const int*)d_in[1];
  const float* ew   = (const float*)d_in[2];
  const float* Wx0  = (const float*)d_in[3];
  const float* Wh0  = (const float*)d_in[4];
  const float* bx0  = (const float*)d_in[5];
  const float* bh0  = (const float*)d_in[6];
  const float* Wx1  = (const float*)d_in[7];
  const float* Wh1  = (const float*)d_in[8];
  const float* bx1  = (const float*)d_in[9];
  const float* bh1  = (const float*)d_in[10];
  const float* muW  = (const float*)d_in[11];
  const float* mub  = (const float*)d_in[12];
  const float* sgW  = (const float*)d_in[13];
  const float* sgb  = (const float*)d_in[14];

  const int E = in_sizes[1] / 2;
  const int* row = eidx;
  const int* col = eidx + E;

  float* ws = (float*)d_ws;
  float* norm = ws;                      ws += E;
  float* deg  = ws;                      ws += Nn;
  float* diag = ws;                      ws += Nn;
  float* Tbuf = ws;                      ws += (size_t)Mrows * 96;
  float* gx   = ws;                      ws += (size_t)Mrows * 96;
  float* HR   = ws;                      ws += (size_t)Mrows * 32;
  float* H1   = ws;                      ws += (size_t)Mrows * 32;
  float* H2   = ws;                      ws += (size_t)Mrows * 32;
  float* Wxp0 = ws;                      ws += 24 * 96;
  float* Wxp1 = ws;                      ws += 96 * 96;
  float* Wzr0 = ws;                      ws += 96 * 64;
  float* W20  = ws;                      ws += 96 * 32;
  float* Wzr1 = ws;                      ws += 96 * 64;
  float* W21  = ws;                      ws += 96 * 32;
  float* mixt = ws;                      ws += 128;
  int2* sedge = (int2*)ws;               ws += 2 * (size_t)E;
  int* cnt  = (int*)ws;                  ws += Nn;
  int* off  = (int*)ws;                  ws += Nn + 1;
  int* cur  = (int*)ws;                  ws += Nn;
  int* bsum = (int*)ws;                  ws += 256;

  auto nblk = [](long n, int b) { return (unsigned)((n + b - 1) / b); };

  // --- graph normalization ---
  k_zero<<<nblk(Nn, 256), 256, 0, stream>>>(deg, Nn);
  k_deg<<<nblk(E, 256), 256, 0, stream>>>(row, ew, deg, E);
  k_dinv_diag<<<nblk(Nn, 256), 256, 0, stream>>>(deg, diag, Nn);
  k_norm<<<nblk(E, 256), 256, 0, stream>>>(row, col, ew, deg, norm, E);

  // --- CSC build (one-time; steady state is atomic-free) ---
  k_zero_i<<<nblk(Nn, 256), 256, 0, stream>>>(cnt, Nn);
  k_count<<<nblk(E, 256), 256, 0, stream>>>(col, cnt, E);
  k_scan_block<<<ScanB, 256, 0, stream>>>(cnt, off, bsum, Nn);
  k_scan_bsums<<<1, 256, 0, stream>>>(bsum, ScanB);
  k_scan_final<<<nblk(Nn, 256), 256, 0, stream>>>(cnt, off, bsum, cur, Nn, E);
  k_place<<<nblk(E, 256), 256, 0, stream>>>(row, col, norm, cur, sedge, E);

  // --- weight repack + state init ---
  k_pack_Wx<<<nblk(3 * Kk * Fin * Hh, 256), 256, 0, stream>>>(Wx0, Wxp0, Fin);
  k_pack_Wx<<<nblk(3 * Kk * Hh * Hh, 256), 256, 0, stream>>>(Wx1, Wxp1, Hh);
  k_pack_Wh<<<nblk(3 * Kk * Hh * Hh, 256), 256, 0, stream>>>(Wh0, Wzr0, W20);
  k_pack_Wh<<<nblk(3 * Kk * Hh * Hh, 256), 256, 0, stream>>>(Wh1, Wzr1, W21);
  k_zero<<<nblk((long)Mrows * 32, 256), 256, 0, stream>>>(H1, (long)Mrows * 32);
  k_zero<<<nblk((long)Mrows * 32, 256), 256, 0, stream>>>(H2, (long)Mrows * 32);

  auto cheb = [&](const float* x, long xbstride, int F) {
    int total = Mrows * F;
    k_copy0<<<nblk(total, 256), 256, 0, stream>>>(x, xbstride, Tbuf, F, total);
    if (F == 8) {
      dim3 gg(nblk(Nn, 32), Bb);
      k_gather<8, 1><<<gg, 256, 0, stream>>>(x, xbstride, Tbuf, sedge, off, diag);
      k_gather<8, 2><<<gg, 256, 0, stream>>>(nullptr, 0, Tbuf, sedge, off, diag);
    } else {
      dim3 gg(nblk(Nn, 8), Bb);
      k_gather<32, 1><<<gg, 256, 0, stream>>>(x, xbstride, Tbuf, sedge, off, diag);
      k_gather<32, 2><<<gg, 256, 0, stream>>>(nullptr, 0, Tbuf, sedge, off, diag);
    }
  };

  auto gru_step = [&](const float* x, long xbstride, int F, const float* Wxp,
                      const float* bx, const float* Wzr, const float* W2,
                      const float* bh, float* H) {
    cheb(x, xbstride, F);                               // TX in Tbuf [M, 3F]
    dim3 g1(nblk(Mrows, 128), 6);
    k_gemm_bias<<<g1, 256, 0, stream>>>(Tbuf, Wxp, bx, gx, Mrows, 3 * F, 96);
    cheb(H, (long)Nn * 32, 32);                         // TH in Tbuf [M, 96]
    dim3 g2(nblk(Mrows, 128), 4);
    k_gates_zr<<<g2, 256, 0, stream>>>(Tbuf, Wzr, bh, gx, H, HR, Mrows);
    cheb(HR, (long)Nn * 32, 32);                        // T(H*R) in Tbuf
    dim3 g3(nblk(Mrows, 128), 2);
    k_gates_h<<<g3, 256, 0, stream>>>(Tbuf, W2, bh, gx, H, Mrows);
  };

  for (int t = 0; t < Tt; ++t) {
    gru_step(in + (size_t)t * Nn * Fin, (long)Tt * Nn * Fin, Fin,
             Wxp0, bx0, Wzr0, W20, bh0, H1);
    gru_step(H1, (long)Nn * 32, Hh, Wxp1, bx1, Wzr1, W21, bh1, H2);
  }

  float* mu  = (float*)d_out;
  float* sg  = mu + (size_t)Mrows * 2;
  float* mix = sg + (size_t)Mrows * 2;
  k_head<<<nblk(Mrows, 256), 256, 0, stream>>>(H2, muW, mub, sgW, sgb, mu, sg, Mrows);
  k_zero<<<1, 128, 0, stream>>>(mixt, 128);
  k_mix_reduce<<<256, 256, 0, stream>>>(H2, mixt, (long)Mrows * 32);
  k_mix_softmax<<<1, 32, 0, stream>>>(mixt, mix);
}